// FlowReversal_55336358643335
// MI455X (gfx1250) — compile-verified
//
#include <hip/hip_runtime.h>

// FlowReversal forward-splat for MI455X (gfx1250, wave32).
// Memory/atomic-bound scatter: no matrix ops exist in this workload. CDNA5
// paths used: no-return GLOBAL_ATOMIC_ADD_F32 (L2-resolved; the 177MB output
// set fits the 192MB L2) + non-temporal b128 loads so the read-once 147MB
// img/flo stream does not evict the atomic-accumulated outputs from L2.

typedef float v4f __attribute__((ext_vector_type(4)));

#define N_   8
#define C_   3
#define H_   720
#define W_   1280
#define HW_  (H_ * W_)          // 921600
#define CHW_ (C_ * HW_)         // 2764800

__global__ __launch_bounds__(256)
void fr_zero_kernel(float* __restrict__ out, int n4) {
    int i = blockIdx.x * blockDim.x + threadIdx.x;
    if (i < n4) {
        v4f z = {0.f, 0.f, 0.f, 0.f};
        ((v4f*)out)[i] = z;
    }
}

__device__ __forceinline__
void fr_atomic_add(float* p, float v) {
    // Relaxed, agent-scope, result discarded -> native global_atomic_add_f32
    // without return (tracked by STOREcnt, RMW performed at L2).
    (void)__hip_atomic_fetch_add(p, v, __ATOMIC_RELAXED, __HIP_MEMORY_SCOPE_AGENT);
}

__device__ __forceinline__
void fr_splat_target(float* __restrict__ imgw_n, float* __restrict__ o_n,
                     int tx, int ty, float wt, float c0, float c1, float c2) {
    // unsigned-compare bounds trick: negative -> huge unsigned -> rejected
    if ((unsigned)tx < (unsigned)H_ && (unsigned)ty < (unsigned)W_) {
        int idx = tx * W_ + ty;
        fr_atomic_add(imgw_n + idx,           c0 * wt);
        fr_atomic_add(imgw_n + HW_ + idx,     c1 * wt);
        fr_atomic_add(imgw_n + 2 * HW_ + idx, c2 * wt);
        fr_atomic_add(o_n + idx, wt);   // channel 0 only; broadcast later
    }
}

// Grid: (1, H, N). Block: 320 threads (10 wave32) = one full image row,
// 4 pixels per thread -> zero integer divisions anywhere.
__global__ __launch_bounds__(320)
void fr_splat_kernel(const float* __restrict__ img,
                     const float* __restrict__ flo,
                     float* __restrict__ imgw,
                     float* __restrict__ o) {
    const int n  = blockIdx.z;
    const int h  = blockIdx.y;
    const int w0 = threadIdx.x * 4;
    const int rowoff = h * W_ + w0;

    // flo[n,0] shifts along W (ref 'y'), flo[n,1] shifts along H (ref 'x').
    const float* floY = flo + (size_t)n * 2 * HW_ + rowoff;
    const float* floX = floY + HW_;
    v4f y4 = __builtin_nontemporal_load((const v4f*)floY);   // global_load_b128 th:NT
    v4f x4 = __builtin_nontemporal_load((const v4f*)floX);

    const float* imgp = img + (size_t)n * CHW_ + rowoff;
    v4f i0 = __builtin_nontemporal_load((const v4f*)(imgp));
    v4f i1 = __builtin_nontemporal_load((const v4f*)(imgp + HW_));
    v4f i2 = __builtin_nontemporal_load((const v4f*)(imgp + 2 * HW_));

    float* imgw_n = imgw + (size_t)n * CHW_;
    float* o_n    = o    + (size_t)n * CHW_;

#pragma unroll
    for (int j = 0; j < 4; ++j) {
        float x = x4[j], y = y4[j];
        float fx = floorf(x), fy = floorf(y);
        float ax = x - fx,   ay = y - fy;         // in [0,1)
        // separable Gaussian: exp(-(dx^2+dy^2)) = exp(-dx^2)*exp(-dy^2)
        float ex1 = __expf(-(ax * ax));
        float ex2 = __expf(-((ax - 1.f) * (ax - 1.f)));
        float ey1 = __expf(-(ay * ay));
        float ey2 = __expf(-((ay - 1.f) * (ay - 1.f)));

        int tx1 = h      + (int)fx;   int tx2 = tx1 + 1;   // H-axis target
        int ty1 = w0 + j + (int)fy;   int ty2 = ty1 + 1;   // W-axis target

        float c0 = i0[j], c1 = i1[j], c2 = i2[j];

        fr_splat_target(imgw_n, o_n, tx1, ty1, ex1 * ey1, c0, c1, c2); // w11
        fr_splat_target(imgw_n, o_n, tx1, ty2, ex1 * ey2, c0, c1, c2); // w12
        fr_splat_target(imgw_n, o_n, tx2, ty1, ex2 * ey1, c0, c1, c2); // w21
        fr_splat_target(imgw_n, o_n, tx2, ty2, ex2 * ey2, c0, c1, c2); // w22
    }
}

// o[n,1,:,:] = o[n,2,:,:] = o[n,0,:,:]  (weights are channel-independent).
// Grid: (1, H, N), block 320, 4 px/thread; reads hit L2 (just accumulated).
__global__ __launch_bounds__(320)
void fr_bcast_kernel(float* __restrict__ o) {
    const int n  = blockIdx.z;
    const int h  = blockIdx.y;
    const int w0 = threadIdx.x * 4;
    float* o_n = o + (size_t)n * CHW_ + h * W_ + w0;
    v4f v = *(const v4f*)o_n;
    *(v4f*)(o_n + HW_)     = v;
    *(v4f*)(o_n + 2 * HW_) = v;
}

extern "C" void kernel_launch(void* const* d_in, const int* in_sizes, int n_in,
                              void* d_out, int out_size, void* d_ws, size_t ws_size,
                              hipStream_t stream) {
    (void)in_sizes; (void)n_in; (void)d_ws; (void)ws_size; (void)out_size;

    const float* img = (const float*)d_in[0];   // (8,3,720,1280) f32
    const float* flo = (const float*)d_in[1];   // (8,2,720,1280) f32
    float* imgw = (float*)d_out;                // first output plane set
    float* o    = imgw + (size_t)N_ * CHW_;     // second output plane set

    // 1) zero both outputs (harness poisons d_out; atomics need zeros)
    int n4 = (2 * N_ * CHW_) / 4;               // 11,059,200 float4 stores
    fr_zero_kernel<<<n4 / 256, 256, 0, stream>>>((float*)d_out, n4);

    // 2) scatter-add splat (atomics into L2-resident outputs)
    dim3 grid(1, H_, N_);                       // one 10-wave block per row
    fr_splat_kernel<<<grid, 320, 0, stream>>>(img, flo, imgw, o);

    // 3) replicate weight-sum to channels 1..2
    fr_bcast_kernel<<<grid, 320, 0, stream>>>(o);
}